// RelationMultiHeadSelfAttention_77249281786232
// MI455X (gfx1250) — compile-verified
//
#include <hip/hip_runtime.h>
#include <hip/hip_bf16.h>
#include <math.h>

typedef __attribute__((ext_vector_type(2))) float v2f;
typedef __attribute__((ext_vector_type(8))) float v8f;

#define NBATCH 4
#define QLEN   256
#define KLEN   256
#define EMBED  256
#define HEADS  8
#define HDIM   32

// Tiling for the fused single-pass relation max-pool
#define QT  16
#define KT  32
#define NQT (QLEN / QT)  // 16
#define NKT (KLEN / KT)  // 8

// ---------------------------------------------------------------------------
// D = A(16x4) * B(4x16) + C  in f32, wave32.
// a: lane m=l&15, half hh=l>>4 -> {A[m, kb+2hh], A[m, kb+2hh+1]}
// b: lane n=l&15, half hh      -> {B[kb+2hh, n], B[kb+2hh+1, n]}
// c: vgpr v, lane l            -> C[v + 8*(l>>4), l&15]
// ---------------------------------------------------------------------------
__device__ __forceinline__ v8f wmma_f32_step(v2f a, v2f b, v8f c) {
  return __builtin_amdgcn_wmma_f32_16x16x4_f32(
      /*neg_a=*/false, a, /*neg_b=*/false, b,
      /*c_mod=*/(short)0, c, /*reuse_a=*/false, /*reuse_b=*/false);
}

// ---------------------------------------------------------------------------
// Single pass over `relation` (268 MB): each block owns a (QT x KT) tile of
// the (q,k) plane for one n, thread = e. Produces BOTH partial reductions:
//   prow[n,q,kt,e]  = max over k in tile   (later reduced over kt)
//   pcol[n,qt,k,e]  = max over q in tile   (later reduced over qt)
// Non-temporal loads: pure stream, keep L2 for the attention working set.
// ---------------------------------------------------------------------------
__global__ __launch_bounds__(256) void relmax_partial_kernel(
    const float* __restrict__ rel, float* __restrict__ prow,
    float* __restrict__ pcol) {
  const int e  = threadIdx.x;
  const int kt = blockIdx.x;  // 8
  const int qt = blockIdx.y;  // 16
  const int n  = blockIdx.z;  // 4
  const int q0 = qt * QT;
  const int k0 = kt * KT;

  float cmax[KT];
#pragma unroll
  for (int k = 0; k < KT; ++k) cmax[k] = -INFINITY;

  for (int q = 0; q < QT; ++q) {
    const float* base =
        rel + ((size_t)(n * QLEN + q0 + q) * KLEN + k0) * EMBED + e;
    float rm = -INFINITY;
#pragma unroll
    for (int k = 0; k < KT; ++k) {
      float v = __builtin_nontemporal_load(base + (size_t)k * EMBED);
      rm = fmaxf(rm, v);
      cmax[k] = fmaxf(cmax[k], v);
    }
    prow[((size_t)(n * QLEN + q0 + q) * NKT + kt) * EMBED + e] = rm;
  }
#pragma unroll
  for (int k = 0; k < KT; ++k)
    pcol[((size_t)(n * NQT + qt) * KLEN + k0 + k) * EMBED + e] = cmax[k];
}

// rel_row[n,q,e] = max over the NKT partials
__global__ __launch_bounds__(256) void rowreduce_kernel(
    const float* __restrict__ prow, float* __restrict__ rel_row) {
  const int e = threadIdx.x, q = blockIdx.x, n = blockIdx.y;
  float m = -INFINITY;
#pragma unroll
  for (int kt = 0; kt < NKT; ++kt)
    m = fmaxf(m, prow[((size_t)(n * QLEN + q) * NKT + kt) * EMBED + e]);
  rel_row[(size_t)(n * QLEN + q) * EMBED + e] = m;
}

// rel_col[n,k,e] = max over the NQT partials
__global__ __launch_bounds__(256) void colreduce_kernel(
    const float* __restrict__ pcol, float* __restrict__ rel_col) {
  const int e = threadIdx.x, k = blockIdx.x, n = blockIdx.y;
  float m = -INFINITY;
#pragma unroll
  for (int qt = 0; qt < NQT; ++qt)
    m = fmaxf(m, pcol[((size_t)(n * NQT + qt) * KLEN + k) * EMBED + e]);
  rel_col[(size_t)(n * KLEN + k) * EMBED + e] = m;
}

// ---------------------------------------------------------------------------
// Per-head projections (torch Linear: y = x @ W.T), fused with rel adds:
//   which==0: v_proj = values @ Wv.T
//   which==1: k_rel  = keys   @ Wk.T + rel_col
//   which==2: q_rel  = query  @ Wq.T + rel_row
// One wave per 16x16 output tile. K-depth = 32 -> 8 WMMA steps.
// ---------------------------------------------------------------------------
__global__ __launch_bounds__(32) void proj_kernel(
    const float* __restrict__ values, const float* __restrict__ keys,
    const float* __restrict__ query,  const float* __restrict__ Wv,
    const float* __restrict__ Wk,     const float* __restrict__ Wq,
    const float* __restrict__ rel_row, const float* __restrict__ rel_col,
    float* __restrict__ v_proj, float* __restrict__ q_rel,
    float* __restrict__ k_rel) {
  const int lane = threadIdx.x;
  const int m  = lane & 15;
  const int hh = lane >> 4;
  const int rt = blockIdx.x;        // 64 row tiles over N*T = 1024 rows
  const int ct = blockIdx.y & 1;    // 2 col tiles over D = 32
  const int h  = blockIdx.y >> 1;   // 8 heads
  const int which = blockIdx.z;     // 0:v 1:k 2:q

  const float* X = (which == 0) ? values : ((which == 1) ? keys : query);
  const float* W = (which == 0) ? Wv     : ((which == 1) ? Wk   : Wq);
  float* dst     = (which == 0) ? v_proj : ((which == 1) ? k_rel : q_rel);

  const int r0 = rt * 16;
  const int e0 = ct * 16;

  v8f c = {};
#pragma unroll
  for (int s = 0; s < 8; ++s) {
    const int kb = 4 * s + 2 * hh;
    v2f a, b;
    a.x = X[(size_t)(r0 + m) * EMBED + h * HDIM + kb + 0];
    a.y = X[(size_t)(r0 + m) * EMBED + h * HDIM + kb + 1];
    b.x = W[(e0 + m) * HDIM + kb + 0];   // B[k,n] = W[e0+n, k], n == m
    b.y = W[(e0 + m) * HDIM + kb + 1];
    c = wmma_f32_step(a, b, c);
  }

#pragma unroll
  for (int v = 0; v < 8; ++v) {
    const int row = r0 + v + 8 * hh;
    const int col = h * HDIM + e0 + m;
    float y = c[v];
    if (which == 2)      y += rel_row[(size_t)row * EMBED + col];
    else if (which == 1) y += rel_col[(size_t)row * EMBED + col];
    dst[(size_t)row * EMBED + col] = y;
  }
}

// ---------------------------------------------------------------------------
// Per (n, h, 16 q-rows): energy strip (16 x 256) via WMMA into LDS,
// then mask -> /sqrt(E) -> softmax over k -> row-sum S -> out_pre = S * v_proj.
// A-fragments (q_rel) are loaded once and reused for all 16 k-tiles.
// ---------------------------------------------------------------------------
__global__ __launch_bounds__(32) void attn_kernel(
    const float* __restrict__ q_rel, const float* __restrict__ k_rel,
    const float* __restrict__ v_proj, const int* __restrict__ mask,
    float* __restrict__ out_pre) {
  __shared__ float etile[16][KLEN + 1];

  const int lane = threadIdx.x;
  const int m  = lane & 15;
  const int hh = lane >> 4;
  const int qt = blockIdx.x;  // 16
  const int h  = blockIdx.y;  // 8
  const int n  = blockIdx.z;  // 4
  const int q0 = qt * 16;

  // Load A fragments (q_rel rows q0..q0+15, head h, 32-deep)
  v2f a[8];
#pragma unroll
  for (int s = 0; s < 8; ++s) {
    const int kb = 4 * s + 2 * hh;
    a[s].x = q_rel[(size_t)(n * QLEN + q0 + m) * EMBED + h * HDIM + kb + 0];
    a[s].y = q_rel[(size_t)(n * QLEN + q0 + m) * EMBED + h * HDIM + kb + 1];
  }

  // 16 k-tiles -> 16x256 energy strip
  for (int kt = 0; kt < 16; ++kt) {
    const int k0 = kt * 16;
    v8f c = {};
#pragma unroll
    for (int s = 0; s < 8; ++s) {
      const int kb = 4 * s + 2 * hh;
      v2f b;
      b.x = k_rel[(size_t)(n * KLEN + k0 + m) * EMBED + h * HDIM + kb + 0];
      b.y = k_rel[(size_t)(n * KLEN + k0 + m) * EMBED + h * HDIM + kb + 1];
      c = wmma_f32_step(a[s], b, c);
    }
#pragma unroll
    for (int v = 0; v < 8; ++v) etile[v + 8 * hh][k0 + m] = c[v];
  }
  __syncthreads();

  const float scale = 1.0f / 16.0f;  // 1/sqrt(EMBED)
  for (int r = 0; r < 16; ++r) {
    const int q = q0 + r;
    float ev[8];
    float mx = -INFINITY;
#pragma unroll
    for (int j = 0; j < 8; ++j) {
      const int k = lane + 32 * j;
      float e = etile[r][k];
      if (mask[(size_t)(n * QLEN + q) * KLEN + k] == 1) e = -1e20f;
      e *= scale;
      ev[j] = e;
      mx = fmaxf(mx, e);
    }
#pragma unroll
    for (int off = 16; off > 0; off >>= 1) mx = fmaxf(mx, __shfl_xor(mx, off, 32));
    float sum = 0.0f;
#pragma unroll
    for (int j = 0; j < 8; ++j) { ev[j] = __expf(ev[j] - mx); sum += ev[j]; }
#pragma unroll
    for (int off = 16; off > 0; off >>= 1) sum += __shfl_xor(sum, off, 32);
    const float inv = 1.0f / sum;
    float ssum = 0.0f;
#pragma unroll
    for (int j = 0; j < 8; ++j) ssum += ev[j] * inv;
#pragma unroll
    for (int off = 16; off > 0; off >>= 1) ssum += __shfl_xor(ssum, off, 32);

    // out[n,q,h,d] = (sum_k attn) * v_proj[n,q,h,d];  lane == d
    const size_t idx = (size_t)(n * QLEN + q) * EMBED + h * HDIM + lane;
    out_pre[idx] = ssum * v_proj[idx];
  }
}

// ---------------------------------------------------------------------------
// d_out = out_pre(1024x256) @ Wo.T(256x256) + bo. One wave per 16x16 tile.
// ---------------------------------------------------------------------------
__global__ __launch_bounds__(32) void final_kernel(
    const float* __restrict__ out_pre, const float* __restrict__ Wo,
    const float* __restrict__ bo, float* __restrict__ out) {
  const int lane = threadIdx.x;
  const int m  = lane & 15;
  const int hh = lane >> 4;
  const int r0 = blockIdx.x * 16;  // 64 tiles
  const int c0 = blockIdx.y * 16;  // 16 tiles

  v8f c = {};
#pragma unroll 8
  for (int s = 0; s < 64; ++s) {
    const int kb = 4 * s + 2 * hh;
    v2f a, b;
    a.x = out_pre[(size_t)(r0 + m) * EMBED + kb + 0];
    a.y = out_pre[(size_t)(r0 + m) * EMBED + kb + 1];
    b.x = Wo[(size_t)(c0 + m) * EMBED + kb + 0];
    b.y = Wo[(size_t)(c0 + m) * EMBED + kb + 1];
    c = wmma_f32_step(a, b, c);
  }
#pragma unroll
  for (int v = 0; v < 8; ++v)
    out[(size_t)(r0 + v + 8 * hh) * EMBED + c0 + m] = c[v] + bo[c0 + m];
}

// ---------------------------------------------------------------------------
extern "C" void kernel_launch(void* const* d_in, const int* in_sizes, int n_in,
                              void* d_out, int out_size, void* d_ws, size_t ws_size,
                              hipStream_t stream) {
  const float* values   = (const float*)d_in[0];
  const float* keys     = (const float*)d_in[1];
  const float* query    = (const float*)d_in[2];
  const int*   mask     = (const int*)  d_in[3];
  const float* relation = (const float*)d_in[4];
  const float* Wv       = (const float*)d_in[5];
  const float* Wk       = (const float*)d_in[6];
  const float* Wq       = (const float*)d_in[7];
  const float* Wo       = (const float*)d_in[8];
  const float* bo       = (const float*)d_in[9];
  float* out = (float*)d_out;

  const size_t TOK = (size_t)NBATCH * QLEN * EMBED;  // 262144 floats
  float* ws      = (float*)d_ws;
  float* rel_row = ws + 0 * TOK;
  float* rel_col = ws + 1 * TOK;
  float* q_rel   = ws + 2 * TOK;
  float* k_rel   = ws + 3 * TOK;
  float* v_proj  = ws + 4 * TOK;
  float* out_pre = ws + 5 * TOK;
  float* prow    = ws + 6 * TOK;                          // N*Q*NKT*E = NKT*TOK = 8*TOK floats
  float* pcol    = prow + (size_t)NBATCH * QLEN * NKT * EMBED;  // after prow

  relmax_partial_kernel<<<dim3(NKT, NQT, NBATCH), 256, 0, stream>>>(
      relation, prow, pcol);
  rowreduce_kernel<<<dim3(QLEN, NBATCH), 256, 0, stream>>>(prow, rel_row);
  colreduce_kernel<<<dim3(KLEN, NBATCH), 256, 0, stream>>>(pcol, rel_col);
  proj_kernel<<<dim3(64, 16, 3), 32, 0, stream>>>(
      values, keys, query, Wv, Wk, Wq, rel_row, rel_col, v_proj, q_rel, k_rel);
  attn_kernel<<<dim3(16, HEADS, NBATCH), 32, 0, stream>>>(
      q_rel, k_rel, v_proj, mask, out_pre);
  final_kernel<<<dim3(64, 16), 32, 0, stream>>>(out_pre, Wo, bo, out);
}